// Radon_3994319585777
// MI455X (gfx1250) — compile-verified
//
#include <hip/hip_runtime.h>
#include <hip/hip_bf16.h>
#include <stdint.h>

// Radon transform: out[b,0,j,a] = sum_i bilinear(img_b; R(theta_a) grid(i,j))
// S=768, A=60, B=8.  Exact pixel mapping (unit-pitch rotated grid):
//   px = 383.5 + c*(j-383.5) + s*(i-383.5)
//   py = 383.5 - s*(j-383.5) + c*(i-383.5)

#define S   768
#define NA  60
#define NB  8
#define TJ  64            // output columns per block
#define TI  32            // output rows (i) per staged tile
#define NT  (S / TI)      // 24 i-tiles
#define BW  76            // LDS tile bounding-box cap (worst rotated extent: sqrt(63^2+31^2)+3 ~ 74)
#define BELEMS (BW * BW)  // 5776 floats per buffer

typedef float    v2f  __attribute__((ext_vector_type(2)));
typedef float    v8f  __attribute__((ext_vector_type(8)));
typedef uint32_t u32x4 __attribute__((ext_vector_type(4)));
typedef int      i32x4 __attribute__((ext_vector_type(4)));
typedef int      i32x8 __attribute__((ext_vector_type(8)));

#if __has_builtin(__builtin_amdgcn_tensor_load_to_lds) && __has_builtin(__builtin_amdgcn_s_wait_tensorcnt)
#define USE_TDM 1
#else
#define USE_TDM 0
#endif

__device__ __forceinline__ int imax(int a, int b) { return a > b ? a : b; }
__device__ __forceinline__ int imin(int a, int b) { return a < b ? a : b; }

struct Rect { int xlo, ylo, w, h; };

// Uniform per (block, i-tile): bounding box of all texels a 64x32 output tile samples,
// clamped to the image, +1 texel safety margin, width forced ODD (LDS bank decorrelation).
__device__ __forceinline__ Rect tile_bbox(float cs, float sn, int j0, int i0) {
  const float CTR = 383.5f;
  const float fj0 = (float)j0 - CTR, fj1 = (float)(j0 + TJ - 1) - CTR;
  const float fi0 = (float)i0 - CTR, fi1 = (float)(i0 + TI - 1) - CTR;
  float pxa = CTR + cs * fj0 + sn * fi0, pxb = CTR + cs * fj0 + sn * fi1;
  float pxc = CTR + cs * fj1 + sn * fi0, pxd = CTR + cs * fj1 + sn * fi1;
  float pya = CTR - sn * fj0 + cs * fi0, pyb = CTR - sn * fj0 + cs * fi1;
  float pyc = CTR - sn * fj1 + cs * fi0, pyd = CTR - sn * fj1 + cs * fi1;
  float pxmin = fminf(fminf(pxa, pxb), fminf(pxc, pxd));
  float pxmax = fmaxf(fmaxf(pxa, pxb), fmaxf(pxc, pxd));
  float pymin = fminf(fminf(pya, pyb), fminf(pyc, pyd));
  float pymax = fmaxf(fmaxf(pya, pyb), fmaxf(pyc, pyd));
  int xlo = imax((int)floorf(pxmin) - 1, 0);
  int xhi = imin((int)floorf(pxmax) + 2, S - 1);
  xhi = imax(xhi, xlo);
  xhi = imin(xhi, xlo + (BW - 2));
  if (((xhi - xlo + 1) & 1) == 0) { if (xhi < S - 1) xhi++; else xlo = imax(xlo - 1, 0); }
  int ylo = imax((int)floorf(pymin) - 1, 0);
  int yhi = imin((int)floorf(pymax) + 2, S - 1);
  yhi = imax(yhi, ylo);
  yhi = imin(yhi, ylo + (BW - 1));
  Rect r; r.xlo = xlo; r.ylo = ylo; r.w = xhi - xlo + 1; r.h = yhi - ylo + 1;
  return r;
}

#if USE_TDM
// One 2-D TDM descriptor: tile (w x h) of 4-byte elements from the 768x768 image
// (row stride 768) into LDS at lds_dst, packed rows (LDS row stride == w, odd).
// Group layouts per CDNA5 ISA 8.3/8.4; trailing groups unused (2-D tile) -> zero.
// 6-arg builtin: (uint32x4 g0, int32x8 g1, int32x4 g2, int32x4 g3, int32x8 g4, i32 cpol)
__device__ __forceinline__ void tdm_load_tile(const float* gsrc, float* lds_dst, int w, int h) {
  uint64_t ga = (uint64_t)(uintptr_t)gsrc;
  uint32_t lo = (uint32_t)(uintptr_t)lds_dst;  // low 32 bits of flat LDS addr == LDS byte offset
  u32x4 g0; i32x8 g1;
  i32x4 g2 = {0, 0, 0, 0}, g3 = {0, 0, 0, 0};
  i32x8 g4 = {0, 0, 0, 0, 0, 0, 0, 0};
  g0[0] = 1u;                                                    // count=1 (user descriptor)
  g0[1] = lo;                                                    // lds_addr
  g0[2] = (uint32_t)ga;                                          // global_addr[31:0]
  g0[3] = (uint32_t)((ga >> 32) & 0x01FFFFFFu) | (2u << 30);     // global_addr[56:32] | type=2
  g1[0] = (int)(2u << 16);                                       // wg_mask=0 | data_size=2 (4B)
  g1[1] = (int)((uint32_t)S << 16);                              // tensor_dim0 low16 (768)
  g1[2] = (int)((uint32_t)S << 16);                              // dim0 hi=0 | tensor_dim1 low16
  g1[3] = (int)((uint32_t)w << 16);                              // dim1 hi=0 | tile_dim0=w
  g1[4] = (int)(uint32_t)h;                                      // tile_dim1=h | tile_dim2=0
  g1[5] = S;                                                     // tensor_dim0_stride low32 (768)
  g1[6] = (int)(((uint32_t)(S * S) & 0xFFFFu) << 16);            // stride hi=0 | dim1_stride low16
  g1[7] = (int)((uint32_t)(S * S) >> 16);                        // dim1_stride[47:16]
  __builtin_amdgcn_tensor_load_to_lds(g0, g1, g2, g3, g4, 0);
}
#endif

__global__ __launch_bounds__(256) void radon_fwd_kernel(const float* __restrict__ img,
                                                        float* __restrict__ out) {
  __shared__ float tile[2][BELEMS];  // 2 x 23.1 KB double buffer

  const int blk = blockIdx.x;
  const int jt  = blk % (S / TJ);
  const int a   = (blk / (S / TJ)) % NA;
  const int b   = blk / ((S / TJ) * NA);

  const float theta = (float)a * 0.052359877559829887f;  // pi/60
  const float cs = cosf(theta);
  const float sn = sinf(theta);
  const float CTR = 383.5f;

  const int tid  = (int)threadIdx.x;
  const int col  = tid & (TJ - 1);   // 0..63  -> output column j0+col
  const int isub = tid >> 6;         // 0..3   -> i-phase within a tile
  const int j0   = jt * TJ;

  const float fj  = (float)(j0 + col) - CTR;
  const float bpx = CTR + cs * fj;   // px = bpx + sn*(i-CTR)
  const float bpy = CTR - sn * fj;   // py = bpy + cs*(i-CTR)

  const float* imgb = img + (size_t)b * S * S;

  // Zero both LDS buffers once: every (med3-clamped) gather below then reads a
  // finite value, so masked-out corners can never inject NaN/garbage.
  {
    float* lds = &tile[0][0];
    for (int idx = tid; idx < 2 * BELEMS; idx += 256) lds[idx] = 0.0f;
  }
  __syncthreads();

  Rect r = tile_bbox(cs, sn, j0, 0);
#if USE_TDM
  if (tid == 0) tdm_load_tile(imgb + (size_t)r.ylo * S + r.xlo, &tile[0][0], r.w, r.h);
#endif

  float acc0 = 0.0f, acc1 = 0.0f;

  for (int it = 0; it < NT; ++it) {
    const int buf = it & 1;
    Rect rn = r;
    if (it + 1 < NT) rn = tile_bbox(cs, sn, j0, (it + 1) * TI);

#if USE_TDM
    // software pipeline: kick next tile's DMA into the other buffer, then
    // wait for the current tile (TDM completes in order per wave).
    if (it + 1 < NT) {
      if (tid == 0) tdm_load_tile(imgb + (size_t)rn.ylo * S + rn.xlo, &tile[buf ^ 1][0], rn.w, rn.h);
      __builtin_amdgcn_s_wait_tensorcnt(1);
    } else {
      __builtin_amdgcn_s_wait_tensorcnt(0);
    }
    __syncthreads();   // publish TDM-written LDS to all waves
#else
    __syncthreads();   // previous compute done before overwrite
    {
      const int n = r.w * r.h;
      const float* src = imgb + (size_t)r.ylo * S + r.xlo;
      for (int idx = tid; idx < n; idx += 256) {
        int y = idx / r.w;
        int x = idx - y * r.w;
        tile[buf][idx] = src[(size_t)y * S + x];
      }
    }
    __syncthreads();
#endif

    const float* Bt = &tile[buf][0];
    const int wln  = r.w;
    const int bofs = r.ylo * wln + r.xlo;         // rect origin in image-index space
    const int ahi  = BELEMS - 2 - wln;            // so a00+wln+1 stays in-buffer
    const int i0 = it * TI;

    // Incremental sample coordinates: i = i0 + isub + 4k, step 4 per k.
    float px = bpx + sn * ((float)(i0 + isub) - CTR);
    float py = bpy + cs * ((float)(i0 + isub) - CTR);
    const float dpx = 4.0f * sn;
    const float dpy = 4.0f * cs;

#pragma unroll
    for (int k = 0; k < TI / 4; ++k) {
      const float x0f = floorf(px), y0f = floorf(py);
      const float wx1 = px - x0f, wy1 = py - y0f;
      const float wx0 = 1.0f - wx1, wy0 = 1.0f - wy1;
      const int ix0 = (int)x0f, iy0 = (int)y0f;
      // Masked weights implement padding_mode='zeros' (per-corner validity).
      const float wx0m = ((unsigned)ix0       < (unsigned)S) ? wx0 : 0.0f;
      const float wx1m = ((unsigned)(ix0 + 1) < (unsigned)S) ? wx1 : 0.0f;
      const float wy0m = ((unsigned)iy0       < (unsigned)S) ? wy0 : 0.0f;
      const float wy1m = ((unsigned)(iy0 + 1) < (unsigned)S) ? wy1 : 0.0f;
      // Any in-image texel lies inside the staged rect; med3-clamp only keeps
      // fully-outside samples (weight 0) from leaving the LDS buffer.
      const int a00 = imin(imax(iy0 * wln + ix0 - bofs, 0), ahi);
      const float v00 = Bt[a00];
      const float v01 = Bt[a00 + 1];          // folds into ds offset
      const float v10 = Bt[a00 + wln];
      const float v11 = Bt[a00 + wln + 1];    // folds into ds offset
      const float t0 = wx0m * v00 + wx1m * v01;
      const float t1 = wx0m * v10 + wx1m * v11;
      if (k & 1) acc1 += wy0m * t0 + wy1m * t1;
      else       acc0 += wy0m * t0 + wy1m * t1;
      px += dpx; py += dpy;
    }

#if USE_TDM
    __syncthreads();   // buffer free before it+2's DMA overwrites it
#endif
    r = rn;
  }

  const float acc = acc0 + acc1;

  // Reduce 4 i-phase partials per column. P[k][n] at tile[0][k*64+n].
  __syncthreads();
  tile[0][tid] = acc;
  __syncthreads();

#if __has_builtin(__builtin_amdgcn_wmma_f32_16x16x4_f32)
  // colsum = ones(16x4) x P(4x16) via one V_WMMA_F32_16X16X4_F32 per 16 columns.
  // B layout (4x16, 32-bit): VGPR0 = {K=0 | K=2}, VGPR1 = {K=1 | K=3} across lane halves.
  // D layout: VGPR0 lanes 0..15 hold row M=0 -> the column sums.
  if (tid < 128) {                    // waves 0..3, full 32-lane EXEC per wave
    const int lane = tid & 31;
    const int n    = ((tid >> 5) << 4) + (lane & 15);  // column within [0,64)
    const int kk   = (lane >> 4) << 1;                 // 0 or 2
    v2f av; av[0] = 1.0f; av[1] = 1.0f;
    v2f bv; bv[0] = tile[0][kk * 64 + n]; bv[1] = tile[0][(kk + 1) * 64 + n];
    v8f cv = {};
    v8f dv = __builtin_amdgcn_wmma_f32_16x16x4_f32(false, av, false, bv,
                                                   (short)0, cv, false, false);
    if (lane < 16)
      out[((size_t)b * S + (size_t)(j0 + n)) * NA + a] = dv[0];
  }
#else
  if (tid < TJ) {
    float s0 = tile[0][tid] + tile[0][64 + tid] + tile[0][128 + tid] + tile[0][192 + tid];
    out[((size_t)b * S + (size_t)(j0 + tid)) * NA + a] = s0;
  }
#endif
}

extern "C" void kernel_launch(void* const* d_in, const int* in_sizes, int n_in,
                              void* d_out, int out_size, void* d_ws, size_t ws_size,
                              hipStream_t stream) {
  (void)in_sizes; (void)n_in; (void)d_ws; (void)ws_size; (void)out_size;
  const float* img = (const float*)d_in[0];
  float* out = (float*)d_out;
  dim3 grid((S / TJ) * NA * NB);   // 12 * 60 * 8 = 5760 blocks
  radon_fwd_kernel<<<grid, 256, 0, stream>>>(img, out);
}